// VisionTransformer_69776038690995
// MI455X (gfx1250) — compile-verified
//
#include <hip/hip_runtime.h>
#include <math.h>

#define LNUM  4
#define EDIM  4
#define HDIM  512
#define NHEAD 8
#define HD    64
#define FF    1024
#define SEQ   1024
#define BATCH 4
#define NT    17
#define MRP   32
#define MTOT  (BATCH*SEQ)   /* 4096 rows */
#define H3    (3*HDIM)      /* 1536 */
#define LNEPS 1e-5f

typedef float v2f __attribute__((ext_vector_type(2)));
typedef float v8f __attribute__((ext_vector_type(8)));

// Full-precision f32 WMMA: D(16x16) = A(16x4) * B(4x16) + C
__device__ __forceinline__ v8f wmma_f32_k4(v2f a, v2f b, v8f c) {
  return __builtin_amdgcn_wmma_f32_16x16x4_f32(false, a, false, b, (short)0, c,
                                               false, false);
}

// ---- CDNA5 async global->LDS copy (ASYNCcnt path) -------------------------
typedef int v4i __attribute__((__vector_size__(4 * sizeof(int))));
typedef __attribute__((address_space(1))) v4i as1_v4i;
typedef __attribute__((address_space(3))) v4i as3_v4i;

__device__ __forceinline__ void async_b128(const float* g, float* l) {
#if __has_builtin(__builtin_amdgcn_global_load_async_to_lds_b128)
  __builtin_amdgcn_global_load_async_to_lds_b128(
      (as1_v4i*)const_cast<float*>(g), (as3_v4i*)l, 0, 0);
#else
  unsigned lo = (unsigned)(size_t)(void*)l;
  unsigned long long ga = (unsigned long long)(size_t)g;
  asm volatile("global_load_async_to_lds_b128 %0, %1, off"
               :: "v"(lo), "v"(ga) : "memory");
#endif
}

__device__ __forceinline__ void wait_async0() {
#if __has_builtin(__builtin_amdgcn_s_wait_asynccnt)
  __builtin_amdgcn_s_wait_asynccnt(0);
#else
  asm volatile("s_wait_asynccnt 0x0" ::: "memory");
#endif
}

__device__ __forceinline__ void wave_lds_fence() {
  __builtin_amdgcn_wave_barrier();
#if __has_builtin(__builtin_amdgcn_s_wait_dscnt)
  __builtin_amdgcn_s_wait_dscnt(0);
#else
  asm volatile("s_wait_dscnt 0x0" ::: "memory");
#endif
  __builtin_amdgcn_wave_barrier();
}

// ---------------------------------------------------------------------------
// x = emb_table[patches]; masked tokens -> emb_table[16]
// ---------------------------------------------------------------------------
__global__ __launch_bounds__(256) void k_embed(const int* __restrict__ patches,
                                               const int* __restrict__ mask,
                                               const float* __restrict__ emb,
                                               float* __restrict__ X) {
  int m = blockIdx.x * 256 + threadIdx.x;
  if (m >= MTOT) return;
  int row = mask[m] ? (NT - 1) : patches[m];
  ((float4*)X)[m] = ((const float4*)emb)[row];
}

// ---------------------------------------------------------------------------
// biasvec[l][p] = sum_d relemb[l][p][d]   (L x 63)
// ---------------------------------------------------------------------------
__global__ __launch_bounds__(256) void k_biasvec(const float* __restrict__ relemb,
                                                 float* __restrict__ BV) {
  int idx = blockIdx.x * 256 + threadIdx.x;
  if (idx >= LNUM * (2 * MRP - 1)) return;
  const float* r = relemb + (size_t)idx * HD;
  float s = 0.f;
  #pragma unroll 8
  for (int d = 0; d < HD; ++d) s += r[d];
  BV[idx] = s;
}

// ---------------------------------------------------------------------------
// xp = x @ Wi + bi (E=4 -> H=512), then LayerNorm over H -> xn
// ---------------------------------------------------------------------------
__global__ __launch_bounds__(512) void k_projln(const float* __restrict__ X,
                                                const float* __restrict__ Wi,
                                                const float* __restrict__ bi,
                                                const float* __restrict__ g,
                                                const float* __restrict__ bn,
                                                float* __restrict__ XN) {
  const int m  = blockIdx.x;
  const int hh = threadIdx.x;
  const float4 xv = ((const float4*)X)[m];
  float v = xv.x * Wi[hh] + xv.y * Wi[HDIM + hh] + xv.z * Wi[2 * HDIM + hh] +
            xv.w * Wi[3 * HDIM + hh] + bi[hh];
  float s = v, s2 = v * v;
  #pragma unroll
  for (int off = 16; off >= 1; off >>= 1) {
    s  += __shfl_xor(s,  off, 32);
    s2 += __shfl_xor(s2, off, 32);
  }
  __shared__ float r1[16], r2[16];
  __shared__ float mv[2];
  const int wid = hh >> 5, ln = hh & 31;
  if (ln == 0) { r1[wid] = s; r2[wid] = s2; }
  __syncthreads();
  if (hh == 0) {
    float a = 0.f, c = 0.f;
    #pragma unroll
    for (int w = 0; w < 16; ++w) { a += r1[w]; c += r2[w]; }
    float mean = a * (1.0f / HDIM);
    float var  = c * (1.0f / HDIM) - mean * mean;
    mv[0] = mean;
    mv[1] = rsqrtf(var + LNEPS);
  }
  __syncthreads();
  XN[(size_t)m * HDIM + hh] = (v - mv[0]) * mv[1] * g[hh] + bn[hh];
}

// ---------------------------------------------------------------------------
// qkv = xn @ Wqkv + bqkv : (4096x512)@(512x1536)
// block = 256 thr (8 waves); async-stage 512x32 weight panel (64 KB) in LDS,
// each wave computes a 16-row m-tile against two 16-col n-tiles.
// ---------------------------------------------------------------------------
__global__ __launch_bounds__(256) void k_qkv(const float* __restrict__ XN,
                                             const float* __restrict__ W,
                                             const float* __restrict__ bias,
                                             float* __restrict__ out) {
  extern __shared__ float wp[];   // [512][32]
  const int tid = threadIdx.x;
  const int n0 = blockIdx.y * 32;
  #pragma unroll
  for (int i = 0; i < 16; ++i) {
    int chunk = i * 256 + tid;            // 4096 b128 chunks
    int k = chunk >> 3, c = (chunk & 7) * 4;
    async_b128(W + (size_t)k * H3 + n0 + c, wp + k * 32 + c);
  }
  wait_async0();
  __syncthreads();

  const int wave = tid >> 5, lane = tid & 31;
  const int half = lane >> 4, lid = lane & 15;
  const int m0 = blockIdx.x * 128 + wave * 16;
  const float* arow = XN + (size_t)(m0 + lid) * HDIM + 2 * half;
  v8f acc0 = {}, acc1 = {};
  #pragma unroll 4
  for (int k = 0; k < HDIM; k += 4) {
    v2f a = *(const v2f*)(arow + k);
    const float* wl = wp + (k + 2 * half) * 32;
    v2f b0, b1;
    b0.x = wl[lid];      b0.y = wl[32 + lid];
    b1.x = wl[16 + lid]; b1.y = wl[48 + lid];
    acc0 = wmma_f32_k4(a, b0, acc0);
    acc1 = wmma_f32_k4(a, b1, acc1);
  }
  const float bb0 = bias[n0 + lid], bb1 = bias[n0 + 16 + lid];
  #pragma unroll
  for (int r = 0; r < 8; ++r) {
    size_t row = (size_t)(m0 + r + 8 * half) * H3;
    out[row + n0 + lid]      = acc0[r] + bb0;
    out[row + n0 + 16 + lid] = acc1[r] + bb1;
  }
}

// ---------------------------------------------------------------------------
// Flash attention per (b, h): 4 waves/block own 4 q-tiles and share a
// double-buffered async K/V tile stream in LDS. Online softmax as before.
// ---------------------------------------------------------------------------
__global__ __launch_bounds__(128) void k_attn(const float* __restrict__ qkv,
                                              const float* __restrict__ bv,
                                              const int* __restrict__ mask,
                                              float* __restrict__ ao) {
  extern __shared__ float smem[];
  float* kbuf = smem;            // [2][16][64]
  float* vbuf = smem + 2048;     // [2][16][64]
  const int tid  = threadIdx.x;
  const int wave = tid >> 5;
  const int lane = tid & 31;
  const int half = lane >> 4, lid = lane & 15;
  float* pl = smem + 4096 + wave * (16 * 17);

  const int q0 = blockIdx.x * 64 + wave * 16;
  const int h  = blockIdx.y;
  const int b  = blockIdx.z;

  const float* base = qkv + (size_t)b * SEQ * H3;
  const float* qp = base + h * HD;
  const float* kp = base + HDIM + h * HD;
  const float* vp = base + 2 * HDIM + h * HD;
  const int* mk = mask + b * SEQ;

  // per-thread async staging role: waves 0-1 copy K, waves 2-3 copy V
  const float* gsrc = ((tid < 64) ? kp : vp);
  float* ldst = ((tid < 64) ? kbuf : vbuf);
  const int t63 = tid & 63;

  // prologue: tile 0
  #pragma unroll
  for (int j = 0; j < 4; ++j) {
    int chunk = t63 + 64 * j;                 // 256 chunks per 16x64 tile
    int row = chunk >> 4, c = (chunk & 15) * 4;
    async_b128(gsrc + (size_t)row * H3 + c, ldst + row * 64 + c);
  }
  wait_async0();
  __syncthreads();

  // cache Q A-fragments (HD=64 -> 16 K-chunks)
  v2f qa[16];
  {
    const float* qrow = qp + (size_t)(q0 + lid) * H3 + 2 * half;
    #pragma unroll
    for (int kk = 0; kk < 16; ++kk) qa[kk] = *(const v2f*)(qrow + kk * 4);
  }

  v8f o0 = {}, o1 = {}, o2 = {}, o3 = {};
  float m_i = -1e30f;  // lanes 0..15 hold rows 0..15
  float l_i = 0.0f;
  const float scale = 0.125f;

  for (int kt = 0; kt < SEQ / 16; ++kt) {
    const int k0 = kt * 16;
    const int cb = (kt & 1) * 1024;
    // issue async copy of next tile into the other buffer
    if (kt + 1 < SEQ / 16) {
      float* nb = ldst + ((kt + 1) & 1) * 1024;
      const float* ng = gsrc + (size_t)(k0 + 16) * H3;
      #pragma unroll
      for (int j = 0; j < 4; ++j) {
        int chunk = t63 + 64 * j;
        int row = chunk >> 4, c = (chunk & 15) * 4;
        async_b128(ng + (size_t)row * H3 + c, nb + row * 64 + c);
      }
    }
    // ---- S = Q K^T from LDS --------------------------------------------
    v8f sc = {};
    const float* kl = kbuf + cb + lid * 64 + 2 * half;
    #pragma unroll
    for (int kk = 0; kk < 16; ++kk) {
      v2f bf = *(const v2f*)(kl + kk * 4);
      sc = wmma_f32_k4(qa[kk], bf, sc);
    }
    // ---- scale + relpos bias + mask (C layout) --------------------------
    const int  kidx = k0 + lid;
    const bool msk  = mk[kidx] != 0;
    #pragma unroll
    for (int r = 0; r < 8; ++r) {
      int qidx = q0 + r + 8 * half;
      int rel  = qidx - kidx + (MRP - 1);
      rel = rel < 0 ? 0 : (rel > 2 * MRP - 2 ? 2 * MRP - 2 : rel);
      float v = sc[r] * scale + bv[rel];
      if (msk && (qidx != kidx)) v = -1e30f;
      sc[r] = v;
    }
    // ---- tile row max ---------------------------------------------------
    float red[8];
    #pragma unroll
    for (int r = 0; r < 8; ++r) {
      float t = sc[r];
      t = fmaxf(t, __shfl_xor(t, 1, 32));
      t = fmaxf(t, __shfl_xor(t, 2, 32));
      t = fmaxf(t, __shfl_xor(t, 4, 32));
      t = fmaxf(t, __shfl_xor(t, 8, 32));
      red[r] = t;  // lanes0-15: row r; lanes16-31: row 8+r
    }
    float tmax = -1e30f;
    #pragma unroll
    for (int r = 0; r < 8; ++r) {
      float a = __shfl(red[r], 0, 32);
      float c = __shfl(red[r], 16, 32);
      if (lane == r)     tmax = a;
      if (lane == r + 8) tmax = c;
    }
    float newm = fmaxf(m_i, tmax);
    // ---- P = exp(S - rowmax) -------------------------------------------
    #pragma unroll
    for (int r = 0; r < 8; ++r) {
      float nm = __shfl(newm, r + 8 * half, 32);
      sc[r] = __expf(sc[r] - nm);
    }
    // ---- tile row sums --------------------------------------------------
    #pragma unroll
    for (int r = 0; r < 8; ++r) {
      float t = sc[r];
      t += __shfl_xor(t, 1, 32);
      t += __shfl_xor(t, 2, 32);
      t += __shfl_xor(t, 4, 32);
      t += __shfl_xor(t, 8, 32);
      red[r] = t;
    }
    float tsum = 0.0f;
    #pragma unroll
    for (int r = 0; r < 8; ++r) {
      float a = __shfl(red[r], 0, 32);
      float c = __shfl(red[r], 16, 32);
      if (lane == r)     tsum = a;
      if (lane == r + 8) tsum = c;
    }
    float f = __expf(m_i - newm);
    l_i = l_i * f + tsum;
    m_i = newm;
    // ---- rescale running O ---------------------------------------------
    #pragma unroll
    for (int r = 0; r < 8; ++r) {
      float fr = __shfl(f, r + 8 * half, 32);
      o0[r] *= fr; o1[r] *= fr; o2[r] *= fr; o3[r] *= fr;
    }
    // ---- stage P via wave-local LDS (C layout -> A layout) --------------
    #pragma unroll
    for (int r = 0; r < 8; ++r)
      pl[(r + 8 * half) * 17 + lid] = sc[r];
    wave_lds_fence();
    // ---- O += P @ V from LDS -------------------------------------------
    #pragma unroll
    for (int kk = 0; kk < 4; ++kk) {
      v2f pa;
      pa.x = pl[lid * 17 + kk * 4 + 2 * half + 0];
      pa.y = pl[lid * 17 + kk * 4 + 2 * half + 1];
      const float* vl = vbuf + cb + (kk * 4 + 2 * half) * 64;
      v2f b0, b1, b2, b3;
      b0.x = vl[lid];      b0.y = vl[64 + lid];
      b1.x = vl[16 + lid]; b1.y = vl[80 + lid];
      b2.x = vl[32 + lid]; b2.y = vl[96 + lid];
      b3.x = vl[48 + lid]; b3.y = vl[112 + lid];
      o0 = wmma_f32_k4(pa, b0, o0);
      o1 = wmma_f32_k4(pa, b1, o1);
      o2 = wmma_f32_k4(pa, b2, o2);
      o3 = wmma_f32_k4(pa, b3, o3);
    }
    // ---- tile handoff: own async copies done, then block-wide sync ------
    wait_async0();
    __syncthreads();
  }
  // ---- finalize: O /= l, store ------------------------------------------
  float linv = 1.0f / l_i;
  #pragma unroll
  for (int r = 0; r < 8; ++r) {
    float li  = __shfl(linv, r + 8 * half, 32);
    int   row = q0 + r + 8 * half;
    float* orow = ao + ((size_t)b * SEQ + row) * HDIM + h * HD;
    orow[lid]      = o0[r] * li;
    orow[16 + lid] = o1[r] * li;
    orow[32 + lid] = o2[r] * li;
    orow[48 + lid] = o3[r] * li;
  }
}

// ---------------------------------------------------------------------------
// x += ao @ Wo + bo : (4096x512)@(512x4). N=4 -> bandwidth bound, VALU.
// ---------------------------------------------------------------------------
__global__ __launch_bounds__(256) void k_oproj(float* __restrict__ X,
                                               const float* __restrict__ AO,
                                               const float* __restrict__ Wo,
                                               const float* __restrict__ bo) {
  int idx = blockIdx.x * 256 + threadIdx.x;
  if (idx >= MTOT * EDIM) return;
  int m = idx >> 2, e = idx & 3;
  const float* ar = AO + (size_t)m * HDIM;
  float acc = 0.f;
  #pragma unroll 8
  for (int k = 0; k < HDIM; ++k) acc += ar[k] * Wo[k * EDIM + e];
  X[idx] += acc + bo[e];
}

// ---------------------------------------------------------------------------
// x += relu(LN_E(x) @ Wf1 + bf1) @ Wf2 + bf2 ; one block (256 thr) per row
// ---------------------------------------------------------------------------
__global__ __launch_bounds__(256) void k_ffn(float* __restrict__ X,
                                             const float* __restrict__ W1,
                                             const float* __restrict__ b1,
                                             const float* __restrict__ W2,
                                             const float* __restrict__ b2,
                                             const float* __restrict__ g,
                                             const float* __restrict__ bn) {
  const int m = blockIdx.x;
  const int t = threadIdx.x;
  const float4 xv = ((const float4*)X)[m];
  float mean = (xv.x + xv.y + xv.z + xv.w) * 0.25f;
  float d0 = xv.x - mean, d1 = xv.y - mean, d2 = xv.z - mean, d3 = xv.w - mean;
  float inv = rsqrtf((d0 * d0 + d1 * d1 + d2 * d2 + d3 * d3) * 0.25f + LNEPS);
  float y0 = d0 * inv * g[0] + bn[0], y1 = d1 * inv * g[1] + bn[1];
  float y2 = d2 * inv * g[2] + bn[2], y3 = d3 * inv * g[3] + bn[3];
  float a0 = 0.f, a1 = 0.f, a2 = 0.f, a3 = 0.f;
  #pragma unroll
  for (int i = 0; i < FF / 256; ++i) {
    int j = i * 256 + t;
    float h1 = y0 * W1[j] + y1 * W1[FF + j] + y2 * W1[2 * FF + j] +
               y3 * W1[3 * FF + j] + b1[j];
    h1 = fmaxf(h1, 0.f);
    const float* w2 = W2 + j * EDIM;
    a0 += h1 * w2[0]; a1 += h1 * w2[1]; a2 += h1 * w2[2]; a3 += h1 * w2[3];
  }
  #pragma unroll
  for (int off = 16; off >= 1; off >>= 1) {
    a0 += __shfl_xor(a0, off, 32); a1 += __shfl_xor(a1, off, 32);
    a2 += __shfl_xor(a2, off, 32); a3 += __shfl_xor(a3, off, 32);
  }
  __shared__ float red[8][4];
  int wid = t >> 5, ln = t & 31;
  if (ln == 0) { red[wid][0] = a0; red[wid][1] = a1; red[wid][2] = a2; red[wid][3] = a3; }
  __syncthreads();
  if (t < 4) {
    float ssum = b2[t];
    #pragma unroll
    for (int w = 0; w < 8; ++w) ssum += red[w][t];
    float xold = (t == 0) ? xv.x : (t == 1) ? xv.y : (t == 2) ? xv.z : xv.w;
    X[m * EDIM + t] = xold + ssum;
  }
}

// ---------------------------------------------------------------------------
// out = x @ Wout + bout : (4096x4)@(4x17)
// ---------------------------------------------------------------------------
__global__ __launch_bounds__(256) void k_out(const float* __restrict__ X,
                                             const float* __restrict__ Wout,
                                             const float* __restrict__ bout,
                                             float* __restrict__ out) {
  int idx = blockIdx.x * 256 + threadIdx.x;
  if (idx >= MTOT * NT) return;
  int m = idx / NT, tt = idx - m * NT;
  const float* xr = X + m * EDIM;
  out[idx] = xr[0] * Wout[tt] + xr[1] * Wout[NT + tt] +
             xr[2] * Wout[2 * NT + tt] + xr[3] * Wout[3 * NT + tt] + bout[tt];
}

// ---------------------------------------------------------------------------
extern "C" void kernel_launch(void* const* d_in, const int* in_sizes, int n_in,
                              void* d_out, int out_size, void* d_ws, size_t ws_size,
                              hipStream_t stream) {
  (void)in_sizes; (void)n_in; (void)out_size; (void)ws_size;
  const int*   patches = (const int*)d_in[0];
  const int*   mask    = (const int*)d_in[1];
  const float* emb     = (const float*)d_in[2];
  const float* Wi      = (const float*)d_in[3];
  const float* bi      = (const float*)d_in[4];
  const float* Wqkv    = (const float*)d_in[5];
  const float* bqkv    = (const float*)d_in[6];
  const float* Wo      = (const float*)d_in[7];
  const float* bo      = (const float*)d_in[8];
  const float* g1      = (const float*)d_in[9];
  const float* b1n     = (const float*)d_in[10];
  const float* Wf1     = (const float*)d_in[11];
  const float* bf1     = (const float*)d_in[12];
  const float* Wf2     = (const float*)d_in[13];
  const float* bf2     = (const float*)d_in[14];
  const float* g2      = (const float*)d_in[15];
  const float* b2n     = (const float*)d_in[16];
  const float* relemb  = (const float*)d_in[17];
  const float* Wout    = (const float*)d_in[18];
  const float* bout    = (const float*)d_in[19];

  float* wsf = (float*)d_ws;
  float* X   = wsf;                          // 4096*4
  float* BV  = wsf + 16384;                  // 4*63
  float* XN  = wsf + 32768;                  // 4096*512
  float* QKV = XN + (size_t)MTOT * HDIM;     // 4096*1536
  float* AO  = QKV + (size_t)MTOT * H3;      // 4096*512

  const int QKV_SMEM  = HDIM * 32 * sizeof(float);            // 64 KB panel
  const int ATTN_SMEM = (4096 + 4 * 16 * 17) * sizeof(float); // ~20.3 KB
  (void)hipFuncSetAttribute((const void*)k_qkv,
                            hipFuncAttributeMaxDynamicSharedMemorySize, QKV_SMEM);
  (void)hipFuncSetAttribute((const void*)k_attn,
                            hipFuncAttributeMaxDynamicSharedMemorySize, ATTN_SMEM);

  k_embed<<<MTOT / 256, 256, 0, stream>>>(patches, mask, emb, X);
  k_biasvec<<<1, 256, 0, stream>>>(relemb, BV);
  for (int l = 0; l < LNUM; ++l) {
    k_projln<<<MTOT, HDIM, 0, stream>>>(X, Wi + l * EDIM * HDIM, bi + l * HDIM,
                                        g1 + l * HDIM, b1n + l * HDIM, XN);
    k_qkv<<<dim3(MTOT / 128, H3 / 32), 256, QKV_SMEM, stream>>>(
        XN, Wqkv + (size_t)l * HDIM * H3, bqkv + l * H3, QKV);
    k_attn<<<dim3(SEQ / 64, NHEAD, BATCH), 128, ATTN_SMEM, stream>>>(
        QKV, BV + l * (2 * MRP - 1), mask, AO);
    k_oproj<<<(MTOT * EDIM) / 256, 256, 0, stream>>>(X, AO, Wo + l * HDIM * EDIM,
                                                     bo + l * EDIM);
    k_ffn<<<MTOT, 256, 0, stream>>>(X, Wf1 + l * EDIM * FF, bf1 + l * FF,
                                    Wf2 + l * FF * EDIM, bf2 + l * EDIM,
                                    g2 + l * EDIM, b2n + l * EDIM);
  }
  k_out<<<(MTOT * NT + 255) / 256, 256, 0, stream>>>(X, Wout, bout, (float*)d_out);
}